// PromptLearner_66125316489726
// MI455X (gfx1250) — compile-verified
//
#include <hip/hip_runtime.h>

#define VOCAB 49408
#define CDIM  768
#define NVERB 117
#define NCTX  16
#define SEQ   77
#define BATCH 16

typedef __attribute__((ext_vector_type(4))) float v4f;

// One wave per output row (b, v, l): async-copy the 3072-byte source row into
// this wave's LDS slice (ASYNCcnt path), then stream it out with non-temporal
// 128-bit stores. 8 waves / 256-thread block; grid covers rows exactly.
__global__ __launch_bounds__(256) void prompts_kernel(
    const int*   __restrict__ nouns_token,    // (BATCH, SEQ)
    const int*   __restrict__ nouns_numbers,  // (BATCH,)
    const float* __restrict__ ctx,            // (NVERB, NCTX, CDIM)
    const float* __restrict__ emb,            // (VOCAB, CDIM)
    float*       __restrict__ out)            // (BATCH, NVERB, SEQ, CDIM)
{
    __shared__ float smem[8 * CDIM];          // 24 KB: one 768-float row per wave

    const int lane = threadIdx.x & 31;
    const int wave = threadIdx.x >> 5;
    const int row  = blockIdx.x * 8 + wave;   // exact: grid*8 == BATCH*NVERB*SEQ

    const int b   = row / (NVERB * SEQ);
    const int rem = row - b * (NVERB * SEQ);
    const int v   = rem / SEQ;
    const int l   = rem - v * SEQ;

    const int  m      = nouns_numbers[b];
    const bool in_ctx = (l > m) && (l <= m + NCTX);

    const float* srcRow;
    if (in_ctx) {
        const int p = l - m - 1;                       // naturally in [0, NCTX)
        srcRow = ctx + ((size_t)v * NCTX + p) * CDIM;
    } else {
        int s = (l <= m) ? l : (l - NCTX);
        s = s < 0 ? 0 : (s > SEQ - 1 ? SEQ - 1 : s);
        const int tok = nouns_token[b * SEQ + s];
        srcRow = emb + (size_t)tok * CDIM;
    }

    const int ldsBaseF = wave * CDIM;                  // float index of wave's slice

    // 6 async b128 loads: each moves 32 lanes x 16 B = 512 B of the row.
#pragma unroll
    for (int k = 0; k < 6; ++k) {
        const int fofs = k * 128 + lane * 4;           // float offset within row
        unsigned ldsAddr =
            (unsigned)(unsigned long long)(const void*)&smem[ldsBaseF + fofs];
        unsigned long long gAddr =
            (unsigned long long)(const void*)(srcRow + fofs);
        asm volatile("global_load_async_to_lds_b128 %0, %1, off"
                     :: "v"(ldsAddr), "v"(gAddr) : "memory");
    }
    // Per-wave ASYNCcnt; each wave only reads its own LDS slice -> no barrier.
    asm volatile("s_wait_asynccnt 0" ::: "memory");

    float* dst = out + (size_t)row * CDIM;
#pragma unroll
    for (int k = 0; k < 6; ++k) {
        const int fofs = k * 128 + lane * 4;
        v4f val = *(const v4f*)&smem[ldsBaseF + fofs];
        __builtin_nontemporal_store(val, (v4f*)&dst[fofs]);   // streaming 443 MB
    }
}

__global__ void concat_token_kernel(
    const int* __restrict__ nouns_token,
    const int* __restrict__ nouns_numbers,
    const int* __restrict__ prefix,       // (NCTX,)
    float*     __restrict__ out_ct)       // (BATCH*SEQ,) as float
{
    const int i = blockIdx.x * blockDim.x + threadIdx.x;
    if (i >= BATCH * SEQ) return;
    const int b = i / SEQ;
    const int l = i - b * SEQ;
    const int m = nouns_numbers[b];
    const bool in_ctx = (l > m) && (l <= m + NCTX);
    int val;
    if (in_ctx) {
        int p = l - m - 1;
        p = p < 0 ? 0 : (p > NCTX - 1 ? NCTX - 1 : p);
        val = prefix[p];
    } else {
        int s = (l <= m) ? l : (l - NCTX);
        s = s < 0 ? 0 : (s > SEQ - 1 ? SEQ - 1 : s);
        val = nouns_token[b * SEQ + s];
    }
    out_ct[i] = (float)val;
}

extern "C" void kernel_launch(void* const* d_in, const int* in_sizes, int n_in,
                              void* d_out, int out_size, void* d_ws, size_t ws_size,
                              hipStream_t stream) {
    const int*   nouns_token   = (const int*)d_in[0];   // (16, 77)
    const int*   nouns_numbers = (const int*)d_in[1];   // (16,)
    const float* ctx           = (const float*)d_in[2]; // (117, 16, 768)
    const float* emb           = (const float*)d_in[3]; // (49408, 768)
    const int*   prefix        = (const int*)d_in[4];   // (1, 16)
    float* out = (float*)d_out;

    const int    n_rows        = BATCH * NVERB * SEQ;               // 144144
    const size_t prompts_elems = (size_t)n_rows * CDIM;             // 110,702,592

    prompts_kernel<<<n_rows / 8, 256, 0, stream>>>(
        nouns_token, nouns_numbers, ctx, emb, out);

    concat_token_kernel<<<(BATCH * SEQ + 255) / 256, 256, 0, stream>>>(
        nouns_token, nouns_numbers, prefix, out + prompts_elems);
}